// GATLayer_19378892439615
// MI455X (gfx1250) — compile-verified
//
#include <hip/hip_runtime.h>
#include <hip/hip_bf16.h>

typedef __attribute__((ext_vector_type(2))) float v2f;
typedef __attribute__((ext_vector_type(8))) float v8f;

#define ALPHA 0.2f
#define B_  32
#define N_  1024
#define D_  64
#define H_  4
#define E_  16

// ---------------------------------------------------------------------------
// Kernel 1: h[bh][n][e] = sum_d input[b][n][d] * W[h][d][e]
// One wave per 16-row tile; V_WMMA_F32_16X16X4_F32, K loop over D=64.
// A layout (16x4 f32): lanes 0-15 -> M=lane, K={0,1}; lanes 16-31 -> M=lane-16, K={2,3}
// B layout (4x16 f32): lanes 0-15 -> N=lane, K={0,1}; lanes 16-31 -> N=lane-16, K={2,3}
// C layout (16x16 f32): VGPR r -> M = r + (lane<16 ? 0 : 8), N = lane%16
// ---------------------------------------------------------------------------
__global__ __launch_bounds__(128) void gat_h_kernel(const float* __restrict__ input,
                                                    const float* __restrict__ W,
                                                    float* __restrict__ h)
{
    const int lane  = threadIdx.x & 31;
    const int wave  = threadIdx.x >> 5;
    const int tile  = blockIdx.x * 4 + wave;          // 8192 tiles total
    const int tilesPerBH = N_ / 16;                   // 64
    const int bh    = tile / tilesPerBH;
    const int i0    = (tile % tilesPerBH) * 16;
    const int b     = bh / H_;
    const int hh    = bh % H_;

    const int m     = lane & 15;
    const int khalf = (lane >> 4) * 2;                // 0 or 2
    const int row   = i0 + m;

    const float* Arow = input + ((size_t)b * N_ + row) * D_;   // + k
    const float* Bcol = W + (size_t)hh * D_ * E_ + m;          // + k*E

    v8f c = {0.f, 0.f, 0.f, 0.f, 0.f, 0.f, 0.f, 0.f};
    #pragma unroll
    for (int k = 0; k < D_; k += 4) {
        const int kk = k + khalf;
        v2f a;
        a.x = Arow[kk];
        a.y = Arow[kk + 1];
        v2f bb;
        bb.x = Bcol[(size_t)kk * E_];
        bb.y = Bcol[(size_t)(kk + 1) * E_];
        c = __builtin_amdgcn_wmma_f32_16x16x4_f32(false, a, false, bb,
                                                  (short)0, c, false, false);
    }

    float* outp = h + ((size_t)bh * N_ + i0) * E_ + m;
    const int mofs = (lane >> 4) * 8;
    #pragma unroll
    for (int r = 0; r < 8; ++r)
        outp[(size_t)(r + mofs) * E_] = c[r];
}

// ---------------------------------------------------------------------------
// Kernel 2: s_src[bh][n] = h[bh][n][:] . a[E:],  s_dst = h . a[:E]
// ---------------------------------------------------------------------------
__global__ __launch_bounds__(256) void gat_scores_kernel(const float* __restrict__ h,
                                                         const float* __restrict__ a,
                                                         float* __restrict__ s_src,
                                                         float* __restrict__ s_dst)
{
    const int idx = blockIdx.x * 256 + threadIdx.x;   // bh*N + n, 131072 total
    const float* hr = h + (size_t)idx * E_;
    float ss = 0.f, sd = 0.f;
    #pragma unroll
    for (int e = 0; e < E_; ++e) {
        const float v = hr[e];
        sd += v * a[e];        // a_dst = a[:E]
        ss += v * a[E_ + e];   // a_src = a[E:]
    }
    s_src[idx] = ss;
    s_dst[idx] = sd;
}

// ---------------------------------------------------------------------------
// Kernel 3: mx[bh] = max_j s_dst[bh][j]
// ---------------------------------------------------------------------------
__global__ __launch_bounds__(256) void gat_max_kernel(const float* __restrict__ s_dst,
                                                      float* __restrict__ mx)
{
    __shared__ float red[256];
    const int bh = blockIdx.x;
    float m = -3.402823466e38f;
    for (int j = threadIdx.x; j < N_; j += 256)
        m = fmaxf(m, s_dst[bh * N_ + j]);
    red[threadIdx.x] = m;
    __syncthreads();
    #pragma unroll
    for (int s = 128; s > 0; s >>= 1) {
        if (threadIdx.x < s)
            red[threadIdx.x] = fmaxf(red[threadIdx.x], red[threadIdx.x + s]);
        __syncthreads();
    }
    if (threadIdx.x == 0) mx[bh] = red[0];
}

// ---------------------------------------------------------------------------
// Kernel 4: fused attention. One wave per 16-row output tile.
//   w_ij = exp(lrelu(s_src_i + s_dst_j) - m_i),  m_i = lrelu(s_src_i + max_j s_dst_j)
//   acc[16x16] += w[16x4] @ h[4x16] via V_WMMA_F32_16X16X4_F32, j loop over 1024
//   out[b, i, h*E + e] = acc / Z_i
// ---------------------------------------------------------------------------
__global__ __launch_bounds__(128) void gat_attn_kernel(const float* __restrict__ h,
                                                       const float* __restrict__ s_src,
                                                       const float* __restrict__ s_dst,
                                                       const float* __restrict__ mx,
                                                       float* __restrict__ out)
{
    __shared__ float sds[N_];
    const int bh   = blockIdx.x;
    const int b    = bh / H_;
    const int hh   = bh % H_;
    const int lane = threadIdx.x & 31;
    const int wave = threadIdx.x >> 5;
    const int i0   = (blockIdx.y * 4 + wave) * 16;

    for (int j = threadIdx.x; j < N_; j += 128)
        sds[j] = s_dst[bh * N_ + j];
    __syncthreads();

    const int   m     = lane & 15;
    const int   khalf = (lane >> 4) * 2;              // lanes 0-15: j%4 in {0,1}; 16-31: {2,3}
    const float si    = s_src[bh * N_ + i0 + m];
    const float t     = si + mx[bh];
    const float mrow  = fmaxf(t, ALPHA * t);          // lrelu(si + max_j sdst_j)

    const float* hb = h + (size_t)bh * N_ * E_ + m;   // + j*E
    v8f   c = {0.f, 0.f, 0.f, 0.f, 0.f, 0.f, 0.f, 0.f};
    float z = 0.f;

    #pragma unroll 4
    for (int jb = 0; jb < N_; jb += 4) {
        const int   j0 = jb + khalf;
        const float t0 = si + sds[j0];
        const float t1 = si + sds[j0 + 1];
        const float w0 = __expf(fmaxf(t0, ALPHA * t0) - mrow);
        const float w1 = __expf(fmaxf(t1, ALPHA * t1) - mrow);
        z += w0 + w1;
        v2f a;  a.x = w0;  a.y = w1;
        v2f bb;                                       // two coalesced rows of h
        bb.x = hb[(size_t)j0 * E_];
        bb.y = hb[(size_t)(j0 + 1) * E_];
        c = __builtin_amdgcn_wmma_f32_16x16x4_f32(false, a, false, bb,
                                                  (short)0, c, false, false);
    }

    // Z_i: lane L and L+16 hold complementary j-subsets for row i0 + (L&15)
    z += __shfl_xor(z, 16, 32);

    // normalize + store: c[r] belongs to row rr = r + (lane<16 ? 0 : 8); Z[rr] lives in lane rr
    float* ob = out + ((size_t)b * N_ + i0) * (H_ * E_) + hh * E_ + m;
    const int mofs = (lane >> 4) * 8;
    #pragma unroll
    for (int r = 0; r < 8; ++r) {
        const float zr = __shfl(z, r + mofs, 32);
        ob[(size_t)(r + mofs) * (H_ * E_)] = c[r] / zr;
    }
}

// ---------------------------------------------------------------------------
extern "C" void kernel_launch(void* const* d_in, const int* in_sizes, int n_in,
                              void* d_out, int out_size, void* d_ws, size_t ws_size,
                              hipStream_t stream) {
    const float* input = (const float*)d_in[0];   // (32,1024,64)
    // d_in[1] = adj : unused by the reference computation
    const float* W     = (const float*)d_in[2];   // (4,64,16)
    const float* a     = (const float*)d_in[3];   // (32,1)
    float*       out   = (float*)d_out;           // (32,1024,64)

    float* ws = (float*)d_ws;
    const size_t hElems = (size_t)B_ * H_ * N_ * E_;   // 2M floats
    const size_t sElems = (size_t)B_ * H_ * N_;        // 128K floats
    float* h     = ws;
    float* s_src = h + hElems;
    float* s_dst = s_src + sElems;
    float* mx    = s_dst + sElems;

    // 1) h = input @ W  (8192 wmma tiles, 4 waves/block)
    gat_h_kernel<<<2048, 128, 0, stream>>>(input, W, h);
    // 2) scores
    gat_scores_kernel<<<(int)(sElems / 256), 256, 0, stream>>>(h, a, s_src, s_dst);
    // 3) per-(b,h) max of s_dst
    gat_max_kernel<<<B_ * H_, 256, 0, stream>>>(s_dst, mx);
    // 4) fused softmax-attention + WMMA aggregation
    gat_attn_kernel<<<dim3(B_ * H_, N_ / 64), 128, 0, stream>>>(h, s_src, s_dst, mx, out);
}